// Attention_15556371546912
// MI455X (gfx1250) — compile-verified
//
#include <hip/hip_runtime.h>
#include <math.h>

// ---------------------------------------------------------------------------
// MI455X (gfx1250) decode-attention step, fp32 end-to-end.
// Bandwidth-bound (~1.2 GB moved): WMMA F32_16X16X4 for all GEMMs (M=16
// batches matches the 16x16 tile), TDM (tensor_load_to_lds) for K/V tile
// staging with double buffering.
// ---------------------------------------------------------------------------

typedef __attribute__((ext_vector_type(2))) float v2f;
typedef __attribute__((ext_vector_type(8))) float v8f;
typedef unsigned u32x4 __attribute__((ext_vector_type(4)));
typedef int      i32x8 __attribute__((ext_vector_type(8)));
typedef int      i32x4 __attribute__((ext_vector_type(4)));

#define BS        16
#define SEQ       4096
#define HIDDEN    4096
#define N_HEADS   32
#define N_KV      8
#define HEAD_DIM  128
#define KCHUNK    4
#define INV_SQRT_HD 0.08838834764831845f  // 1/sqrt(128)

#if defined(__has_builtin)
#if __has_builtin(__builtin_amdgcn_tensor_load_to_lds) && \
    __has_builtin(__builtin_amdgcn_s_wait_tensorcnt)
#define HAVE_TDM 1
#endif
#endif

__device__ __forceinline__ v8f wmma4(v2f a, v2f b, v8f c) {
  // V_WMMA_F32_16X16X4_F32: D(16x16 f32) = A(16x4 f32) x B(4x16 f32) + C
  return __builtin_amdgcn_wmma_f32_16x16x4_f32(false, a, false, b, (short)0, c,
                                               false, false);
}

#if HAVE_TDM
__device__ __forceinline__ unsigned lds_byte_offset(const void* p) {
  // generic -> LDS(as3) addrspacecast; as3 pointers are 32-bit LDS offsets.
  return (unsigned)(unsigned long long)(__attribute__((address_space(3))) const void*)p;
}

// One TDM descriptor moves a 16x128 fp32 tile (row stride 1024 elements) into
// LDS with 4-DWORD padding after every 128 DWORDs -> padded 132-float rows.
__device__ __forceinline__ void stage_tile(const float* __restrict__ gRow,
                                           float* lbuf, int /*lane*/) {
  const unsigned long long ga = (unsigned long long)gRow;
  u32x4 g0;
  g0.x = 1u;                                   // count=1, user descriptor
  g0.y = lds_byte_offset(lbuf);                // lds_addr (bytes)
  g0.z = (unsigned)(ga & 0xFFFFFFFFu);         // global_addr[31:0]
  g0.w = (unsigned)((ga >> 32) & 0x01FFFFFFu)  // global_addr[56:32]
         | 0x80000000u;                        // type=2 ("image") at [127:126]
  i32x8 g1;
  g1[0] = (2 << 16)      // data_size = 4 bytes
        | (1 << 20)      // pad_enable
        | (6 << 22)      // pad_interval: 128 DWORDs
        | (3 << 25);     // pad_amount:   4 DWORDs
  g1[1] = (128 << 16);   // tensor_dim0[15:0]=128 (bits 63:48); atomic addr=0
  g1[2] = (16 << 16);    // tensor_dim0 hi=0; tensor_dim1[15:0]=16
  g1[3] = (128 << 16);   // tensor_dim1 hi=0; tile_dim0=128
  g1[4] = 16;            // tile_dim1=16; tile_dim2=0
  g1[5] = 1024;          // tensor_dim0_stride = 1024 elements (lo32)
  g1[6] = 0;             // stride hi, tensor_dim1_stride lo
  g1[7] = 0;
  const i32x4 z4 = {0, 0, 0, 0};
  const i32x8 z8 = {0, 0, 0, 0, 0, 0, 0, 0};
  __builtin_amdgcn_tensor_load_to_lds(g0, g1, z4, z4, z8, 0);
}
template <int N>
__device__ __forceinline__ void wait_stage() {
  __builtin_amdgcn_s_wait_tensorcnt(N);
}
__device__ __forceinline__ void guard_lds_reuse() {
  // TDM LDS writes are not ordered with this wave's DS queue: make sure all
  // our outstanding LDS reads finished before the DMA may overwrite a buffer.
  asm volatile("s_wait_dscnt 0x0" ::: "memory");
}
#else
__device__ __forceinline__ void stage_tile(const float* __restrict__ gRow,
                                           float* lbuf, int lane) {
  float4 tmp[16];
#pragma unroll
  for (int r = 0; r < 16; ++r)   // all 16 loads in flight before any LDS store
    tmp[r] = *(const float4*)(gRow + (size_t)r * (N_KV * HEAD_DIM) + lane * 4);
#pragma unroll
  for (int r = 0; r < 16; ++r)
    *(float4*)(lbuf + r * 132 + lane * 4) = tmp[r];
}
template <int N>
__device__ __forceinline__ void wait_stage() {}
__device__ __forceinline__ void guard_lds_reuse() {}
#endif

// ---------------------------------------------------------------------------
// Partial GEMM: C[16, ncols] += A[16, K] x B[K, LDB-major], K split in KCHUNK
// chunks (blockIdx.y) written to separate partial buffers (deterministic).
// One wave computes one 16x16 tile over its K chunk; compile-time strides
// keep the streaming loop pure pointer-increment + immediate-offset loads.
// ---------------------------------------------------------------------------
template <int K, int LDB, int PLD>
__global__ void gemm16_partial(const float* __restrict__ A,
                               const float* __restrict__ B,
                               float* __restrict__ part,
                               int colOffset, int ncols) {
  const int lane = threadIdx.x & 31;
  const int wid  = threadIdx.x >> 5;
  const int tile = blockIdx.x * (blockDim.x >> 5) + wid;
  const int n0   = tile * 16;
  if (n0 >= ncols) return;  // wave-uniform: EXEC stays all-ones for WMMA

  const int nl   = lane & 15;
  const int half = lane >> 4;
  constexpr int KLEN = K / KCHUNK;
  const int kBase = blockIdx.y * KLEN;

  const float* aPtr = A + (size_t)nl * K + kBase + half * 2;
  const float* bPtr = B + (size_t)(kBase + half * 2) * LDB + n0 + nl;

  v8f c = {};
#pragma unroll 4
  for (int k = 0; k < KLEN; k += 4) {
    v2f a;
    a.x = aPtr[0];
    a.y = aPtr[1];
    v2f b;
    b.x = bPtr[0];
    b.y = bPtr[LDB];
    c = wmma4(a, b, c);
    aPtr += 4;
    bPtr += (size_t)4 * LDB;
  }
  // C layout: VGPR i, lanes 0-15 -> (m=i, n), lanes 16-31 -> (m=8+i, n)
  float* dst = part + (size_t)blockIdx.y * 16 * PLD + colOffset + n0 + nl +
               (size_t)half * 8 * PLD;
#pragma unroll
  for (int i = 0; i < 8; ++i) {
    dst[(size_t)i * PLD] = c[i];
  }
}

// ---------------------------------------------------------------------------
// Reduce 4 K-chunk partials of the fused QKV GEMM, apply RoPE, scatter k/v
// into the updated caches at last_pos[b], write q to q_buf.
// Columns: [0,4096)=q, [4096,5120)=k, [5120,6144)=v. One thread = one pair.
// ---------------------------------------------------------------------------
__global__ void qkv_reduce_rope(const float* __restrict__ part,   // [4][16][6144]
                                const float* __restrict__ rope,   // [4096][64][2]
                                const int* __restrict__ last_pos,
                                float* __restrict__ q_buf,        // [16][4096]
                                float* __restrict__ upd_k,
                                float* __restrict__ upd_v) {
  const int idx = blockIdx.x * blockDim.x + threadIdx.x;
  if (idx >= BS * 3072) return;
  const int b   = idx / 3072;
  const int col = (idx % 3072) * 2;
  const size_t o = (size_t)b * 6144 + col;
  float t0 = 0.f, t1 = 0.f;
#pragma unroll
  for (int p = 0; p < KCHUNK; ++p) {
    t0 += part[(size_t)p * (16 * 6144) + o];
    t1 += part[(size_t)p * (16 * 6144) + o + 1];
  }
  const int pos = last_pos[b];
  if (col < 5120) {                       // RoPE on q and k
    const int d = col & 127;              // even; pair index j = d/2
    const float cc = rope[(size_t)pos * HEAD_DIM + d];
    const float ss = rope[(size_t)pos * HEAD_DIM + d + 1];
    const float o0 = t0 * cc - t1 * ss;
    const float o1 = t1 * cc + t0 * ss;
    t0 = o0; t1 = o1;
  }
  if (col < 4096) {
    q_buf[(size_t)b * 4096 + col]     = t0;
    q_buf[(size_t)b * 4096 + col + 1] = t1;
  } else if (col < 5120) {
    const size_t off = ((size_t)b * SEQ + pos) * (N_KV * HEAD_DIM) + (col - 4096);
    upd_k[off] = t0; upd_k[off + 1] = t1;
  } else {
    const size_t off = ((size_t)b * SEQ + pos) * (N_KV * HEAD_DIM) + (col - 5120);
    upd_v[off] = t0; upd_v[off + 1] = t1;
  }
}

// ---------------------------------------------------------------------------
// Flash-attention partial: block = (b, kv-head, 256-seq chunk), 2 waves.
// Each wave handles 128 seq positions (8 tiles of 16) with double-buffered
// TDM staging of K/V tiles into padded LDS (no cross-wave barriers needed:
// tile/P buffers are wave-private, DS is in-order per wave).
// Pass 1: scores via WMMA. q_lds is padded to 16 rows with rows 4-15 zeroed
// once so A-operand loads are unconditional ds_load_b64 (no EXEC masking).
// Pass 2: P = exp(s - m) transposed through LDS to A-layout; O += P x V.
// Writes per-wave partials (m[4], l[4], o[4][128]).
// ---------------------------------------------------------------------------
#define ATT_WAVES 2
__global__ void attn_partial(const float* __restrict__ q_buf,
                             const float* __restrict__ upd_k,
                             const float* __restrict__ upd_v,
                             const float* __restrict__ mask,    // [16][4096]
                             float* __restrict__ att_m,
                             float* __restrict__ att_l,
                             float* __restrict__ att_o) {
  __shared__ __align__(16) float q_lds[16 * 130];                   // [h][d]
  __shared__ __align__(16) float tile_lds[ATT_WAVES][2][16 * 132];  // dbl-buf
  __shared__ __align__(16) float p_lds[ATT_WAVES][16 * 17];         // [s][h]

  const int b  = blockIdx.x;
  const int kh = blockIdx.y;
  const int cz = blockIdx.z;            // 0..15, 256 seq each
  const int tid  = threadIdx.x;
  const int lane = tid & 31;
  const int wid  = tid >> 5;
  const int nl   = lane & 15;
  const int half = lane >> 4;

  // Stage this (b,kh)'s 4 query heads into q_lds rows 0-3; rows 4-15 zero so
  // WMMA A-rows >= 4 are deterministic without per-load predication.
  // Same for the P buffers (rows h>=4 stay zero forever).
  for (int i = tid; i < 16 * 128; i += 64) {
    const int row = i >> 7, col = i & 127;
    q_lds[row * 130 + col] =
        (row < 4)
            ? q_buf[(size_t)b * 4096 + (size_t)kh * 4 * HEAD_DIM + i]
            : 0.f;
  }
  for (int i = tid; i < ATT_WAVES * 16 * 17; i += 64)
    (&p_lds[0][0])[i] = 0.f;
  __syncthreads();

  const int sWave = cz * 256 + wid * 128;
  const float* kBase =
      upd_k + (((size_t)b * SEQ + sWave) * N_KV + kh) * HEAD_DIM;
  const float* vBase =
      upd_v + (((size_t)b * SEQ + sWave) * N_KV + kh) * HEAD_DIM;

  float pm[8][4];
  float vmax[4] = {-1e30f, -1e30f, -1e30f, -1e30f};

  // ---- Pass 1: scores for 8 tiles of 16 seq positions ----
  stage_tile(kBase, tile_lds[wid][0], lane);
#pragma unroll
  for (int t = 0; t < 8; ++t) {
    if (t < 7) {
      guard_lds_reuse();
      stage_tile(kBase + (size_t)(t + 1) * 16 * (N_KV * HEAD_DIM),
                 tile_lds[wid][(t + 1) & 1], lane);
      wait_stage<1>();
    } else {
      wait_stage<0>();
    }
    const float* tl = tile_lds[wid][t & 1];
    v8f c = {};
#pragma unroll
    for (int k0 = 0; k0 < HEAD_DIM; k0 += 4) {
      const int kk = k0 + half * 2;
      v2f a;                                   // A: q (rows 4-15 are zero)
      a.x = q_lds[nl * 130 + kk];
      a.y = q_lds[nl * 130 + kk + 1];
      v2f bb;                                  // B[k][s] = K[s][k]
      bb.x = tl[nl * 132 + kk];
      bb.y = tl[nl * 132 + kk + 1];
      c = wmma4(a, bb, c);
    }
    const float msk = mask[(size_t)b * SEQ + sWave + t * 16 + nl];
#pragma unroll
    for (int i = 0; i < 4; ++i) {
      const float s = c[i] * INV_SQRT_HD + msk;
      pm[t][i] = s;
      vmax[i] = fmaxf(vmax[i], s);
    }
  }

  // Exact max over this wave's 128 positions (reduce across the 16-lane group).
#pragma unroll
  for (int i = 0; i < 4; ++i) {
    float m = vmax[i];
    for (int off = 1; off < 16; off <<= 1) m = fmaxf(m, __shfl_xor(m, off, 32));
    vmax[i] = m;
  }
  float lsum[4] = {0.f, 0.f, 0.f, 0.f};
#pragma unroll
  for (int t = 0; t < 8; ++t)
#pragma unroll
    for (int i = 0; i < 4; ++i) {
      const float e = __expf(pm[t][i] - vmax[i]);
      pm[t][i] = e;
      lsum[i] += e;
    }
#pragma unroll
  for (int i = 0; i < 4; ++i) {
    float s = lsum[i];
    for (int off = 1; off < 16; off <<= 1) s += __shfl_xor(s, off, 32);
    lsum[i] = s;
  }

  // ---- Pass 2: O += P x V via WMMA ----
  v8f acc[8];
#pragma unroll
  for (int n = 0; n < 8; ++n) acc[n] = (v8f){};

  guard_lds_reuse();
  stage_tile(vBase, tile_lds[wid][0], lane);
#pragma unroll
  for (int t = 0; t < 8; ++t) {
    // P (C-layout, lanes 0-15 valid) -> LDS [s][h] for A-layout reads.
    if (half == 0) {
#pragma unroll
      for (int i = 0; i < 4; ++i) p_lds[wid][nl * 17 + i] = pm[t][i];
    }
    if (t < 7) {
      guard_lds_reuse();
      stage_tile(vBase + (size_t)(t + 1) * 16 * (N_KV * HEAD_DIM),
                 tile_lds[wid][(t + 1) & 1], lane);
      wait_stage<1>();
    } else {
      wait_stage<0>();
    }
    const float* tl = tile_lds[wid][t & 1];
#pragma unroll
    for (int kk = 0; kk < 4; ++kk) {          // 4 seq positions per WMMA step
      const int s0 = kk * 4 + half * 2;
      v2f a;                                   // A[h][s] from p_lds[s][h]
      a.x = p_lds[wid][s0 * 17 + nl];
      a.y = p_lds[wid][(s0 + 1) * 17 + nl];
#pragma unroll
      for (int n = 0; n < 8; ++n) {            // 8 x 16 dims of HEAD_DIM
        v2f bb;                                // B[s][d] = V[s][d]
        bb.x = tl[s0 * 132 + n * 16 + nl];
        bb.y = tl[(s0 + 1) * 132 + n * 16 + nl];
        acc[n] = wmma4(a, bb, acc[n]);
      }
    }
  }

  // ---- Write per-wave partials ----
  const int p = cz * ATT_WAVES + wid;                       // 0..31
  const size_t pb = (((size_t)b * N_KV + kh) * 32 + p) * 4;
  if (lane == 0) {
#pragma unroll
    for (int i = 0; i < 4; ++i) {
      att_m[pb + i] = vmax[i];
      att_l[pb + i] = lsum[i];
    }
  }
  if (half == 0) {
#pragma unroll
    for (int i = 0; i < 4; ++i)
#pragma unroll
      for (int n = 0; n < 8; ++n)
        att_o[(pb + i) * HEAD_DIM + n * 16 + nl] = acc[n][i];
  }
}

// ---------------------------------------------------------------------------
// Deterministic sequential merge of the 32 softmax partials per (b, kh).
// ---------------------------------------------------------------------------
__global__ void attn_combine(const float* __restrict__ att_m,
                             const float* __restrict__ att_l,
                             const float* __restrict__ att_o,
                             float* __restrict__ o_buf) {     // [16][4096]
  const int b  = blockIdx.x;
  const int kh = blockIdx.y;
  const int d  = threadIdx.x;                 // 0..127
  const size_t base = ((size_t)b * N_KV + kh) * 32;
  for (int r = 0; r < 4; ++r) {
    float M = -1e30f;
    for (int p = 0; p < 32; ++p) M = fmaxf(M, att_m[(base + p) * 4 + r]);
    float L = 0.f, o = 0.f;
    for (int p = 0; p < 32; ++p) {
      const float w = __expf(att_m[(base + p) * 4 + r] - M);
      L += att_l[(base + p) * 4 + r] * w;
      o += w * att_o[((base + p) * 4 + r) * HEAD_DIM + d];
    }
    o_buf[((size_t)b * N_HEADS + kh * 4 + r) * HEAD_DIM + d] = o / L;
  }
}

__global__ void reduce4(const float* __restrict__ part, float* __restrict__ out,
                        int n) {
  const int i = blockIdx.x * blockDim.x + threadIdx.x;
  if (i < n)
    out[i] = part[i] + part[(size_t)n + i] + part[(size_t)2 * n + i] +
             part[(size_t)3 * n + i];
}

// ---------------------------------------------------------------------------
extern "C" void kernel_launch(void* const* d_in, const int* in_sizes, int n_in,
                              void* d_out, int out_size, void* d_ws,
                              size_t ws_size, hipStream_t stream) {
  (void)in_sizes; (void)n_in; (void)out_size; (void)ws_size;
  const float* x    = (const float*)d_in[0];
  const float* mask = (const float*)d_in[1];
  const float* rope = (const float*)d_in[2];
  const float* wq   = (const float*)d_in[3];
  const float* wk   = (const float*)d_in[4];
  const float* wv   = (const float*)d_in[5];
  const float* wo   = (const float*)d_in[6];
  const float* ck   = (const float*)d_in[7];
  const float* cv   = (const float*)d_in[8];
  const int*   lpos = (const int*)d_in[9];

  float* out_f = (float*)d_out;                         // [16][4096]
  float* upd_k = out_f + (size_t)BS * HIDDEN;           // [16][4096][8][128]
  float* upd_v = upd_k + (size_t)BS * SEQ * N_KV * HEAD_DIM;

  float* ws      = (float*)d_ws;
  float* partQKV = ws;                                  // 4*16*6144
  float* q_buf   = partQKV + 4 * 16 * 6144;             // 16*4096
  float* att_m   = q_buf + BS * HIDDEN;                 // 16*8*32*4
  float* att_l   = att_m + BS * N_KV * 32 * 4;
  float* att_o   = att_l + BS * N_KV * 32 * 4;          // 16*8*32*4*128
  float* o_buf   = att_o + (size_t)BS * N_KV * 32 * 4 * HEAD_DIM;
  float* partOut = o_buf + BS * HIDDEN;                 // 4*16*4096

  // 1) Bulk cache copy (dominant traffic, driver-optimal D2D, capture-safe).
  const size_t cacheBytes = (size_t)BS * SEQ * N_KV * HEAD_DIM * sizeof(float);
  (void)hipMemcpyAsync(upd_k, ck, cacheBytes, hipMemcpyDeviceToDevice, stream);
  (void)hipMemcpyAsync(upd_v, cv, cacheBytes, hipMemcpyDeviceToDevice, stream);

  // 2) Fused QKV projection (K split 4-way for parallelism).
  gemm16_partial<HIDDEN, 4096, 6144>
      <<<dim3(64, KCHUNK), 128, 0, stream>>>(x, wq, partQKV, 0, 4096);
  gemm16_partial<HIDDEN, 1024, 6144>
      <<<dim3(16, KCHUNK), 128, 0, stream>>>(x, wk, partQKV, 4096, 1024);
  gemm16_partial<HIDDEN, 1024, 6144>
      <<<dim3(16, KCHUNK), 128, 0, stream>>>(x, wv, partQKV, 5120, 1024);

  // 3) Reduce + RoPE + scatter into caches.
  qkv_reduce_rope<<<192, 256, 0, stream>>>(partQKV, rope, lpos, q_buf, upd_k,
                                           upd_v);

  // 4) Attention (flash partials + deterministic combine).
  attn_partial<<<dim3(BS, N_KV, 16), 64, 0, stream>>>(q_buf, upd_k, upd_v, mask,
                                                      att_m, att_l, att_o);
  attn_combine<<<dim3(BS, N_KV), 128, 0, stream>>>(att_m, att_l, att_o, o_buf);

  // 5) Output projection.
  gemm16_partial<HIDDEN, 4096, 4096>
      <<<dim3(64, KCHUNK), 128, 0, stream>>>(o_buf, wo, partOut, 0, 4096);
  reduce4<<<256, 256, 0, stream>>>(partOut, out_f, BS * HIDDEN);
}